// LocallyConnectedAutoencoder_38895223832610
// MI455X (gfx1250) — compile-verified
//
#include <hip/hip_runtime.h>

typedef float v2f __attribute__((ext_vector_type(2)));
typedef float v4f __attribute__((ext_vector_type(4)));
typedef float v8f __attribute__((ext_vector_type(8)));

#define HH   256
#define WW   128
#define PS   16
#define NPW  8
#define PNUM 128
#define DD   256
#define HID  8
#define IMG  (HH * WW)       /* 32768 */
#define XT_STRIDE 260        /* 256 + 4 pad -> conflict-free LDS reads */

__global__ __launch_bounds__(32)
void lca_wmma_kernel(const float* __restrict__ x,
                     const float* __restrict__ We,
                     const float* __restrict__ be,
                     const float* __restrict__ Wd,
                     const float* __restrict__ bd,
                     float* __restrict__ out)
{
    __shared__ float xt[16 * XT_STRIDE];   // 16 samples x 256 feats (padded)
    __shared__ float zsh[16 * HID];        // 16 samples x 8 hidden

    const int p    = blockIdx.x;           // patch id
    const int n0   = blockIdx.y * 16;      // sample tile base
    const int ph   = p >> 3;               // patch row  (NPH=16)
    const int pw   = p & (NPW - 1);        // patch col  (NPW=8)
    const int tid  = threadIdx.x;          // 0..31 (one wave)
    const int nn   = tid & 15;             // M (A) / N (B,C) index
    const int hi   = tid >> 4;             // lane half
    const int koff = hi * 2;               // K sub-offset for A/B frags
    const int pimg = ph * (PS * WW) + pw * PS;  // patch origin in image

    // ---- stage 16x256 x-tile into LDS (float4, cacheline-granular) ----
    #pragma unroll
    for (int it = 0; it < 32; ++it) {
        int i   = it * 32 + tid;     // 1024 float4 slots
        int m   = i >> 6;            // sample in tile
        int rem = i & 63;            // float4 within patch
        int r   = rem >> 2;          // patch row
        int c4  = (rem & 3) * 4;     // patch col base
        v4f v = *(const v4f*)(x + (size_t)(n0 + m) * IMG + pimg + r * WW + c4);
        *(v4f*)&xt[m * XT_STRIDE + rem * 4] = v;
    }
    __syncthreads();

    const float* Wep = We + p * (HID * DD);   // [8][256]
    const float* Wdp = Wd + p * (DD * HID);   // [256][8]

    // ---- encode: z[16x8] = X[16x256] @ We_p^T, 64 x v_wmma_f32_16x16x4_f32
    const float bmask = (nn < HID) ? 1.0f : 0.0f;
    const int   bn    = nn & (HID - 1);       // safe address for masked lanes
    v8f acc = {0.f, 0.f, 0.f, 0.f, 0.f, 0.f, 0.f, 0.f};
    #pragma unroll 8
    for (int k0 = 0; k0 < DD; k0 += 4) {
        v2f a = *(const v2f*)&xt[nn * XT_STRIDE + k0 + koff];       // A[m=nn][k]
        v2f b = *(const v2f*)(Wep + bn * DD + k0 + koff);           // B[k][n=nn]
        b.x *= bmask;  b.y *= bmask;
        acc = __builtin_amdgcn_wmma_f32_16x16x4_f32(
                  false, a, false, b, (short)0, acc, false, false);
    }

    // ---- z (+bias) to LDS: convert C-layout -> A-layout source ----
    if (nn < HID) {
        float bev = be[p * HID + nn];
        #pragma unroll
        for (int j = 0; j < 8; ++j) {
            int m = hi ? (j + 8) : j;        // C layout: VGPR j -> M=j / j+8
            zsh[m * HID + nn] = acc[j] + bev;
        }
    }
    __syncthreads();

    // decode A-frags (reused across all 16 output tiles): A[m][h]
    v2f a0 = *(const v2f*)&zsh[nn * HID + koff];        // h = koff,   koff+1
    v2f a1 = *(const v2f*)&zsh[nn * HID + 4 + koff];    // h = 4+koff, 4+koff+1

    // ---- decode: dec[16x256] = Z[16x8] @ Wd_p^T, 16 tiles x 2 WMMA ----
    #pragma unroll 4
    for (int t = 0; t < 16; ++t) {
        int dcol = t * 16 + nn;                          // output feature
        v2f b0 = *(const v2f*)(Wdp + dcol * HID + koff);     // B[h][n]=Wd[p][d][h]
        v2f b1 = *(const v2f*)(Wdp + dcol * HID + 4 + koff);
        v8f dacc = {0.f, 0.f, 0.f, 0.f, 0.f, 0.f, 0.f, 0.f};
        dacc = __builtin_amdgcn_wmma_f32_16x16x4_f32(
                   false, a0, false, b0, (short)0, dacc, false, false);
        dacc = __builtin_amdgcn_wmma_f32_16x16x4_f32(
                   false, a1, false, b1, (short)0, dacc, false, false);

        float bdv = bd[p * DD + dcol];
        int colbase = pimg + t * WW + nn;                // d>>4 = t, d&15 = nn
        #pragma unroll
        for (int j = 0; j < 8; ++j) {
            int m = hi ? (j + 8) : j;
            float v = dacc[j] + bdv;
            out[(size_t)(n0 + m) * IMG + colbase] = 1.0f / (1.0f + __expf(-v));
        }
    }
}

extern "C" void kernel_launch(void* const* d_in, const int* in_sizes, int n_in,
                              void* d_out, int out_size, void* d_ws, size_t ws_size,
                              hipStream_t stream)
{
    (void)n_in; (void)out_size; (void)d_ws; (void)ws_size;
    const float* x  = (const float*)d_in[0];
    const float* We = (const float*)d_in[1];
    const float* be = (const float*)d_in[2];
    const float* Wd = (const float*)d_in[3];
    const float* bd = (const float*)d_in[4];
    float* out = (float*)d_out;

    const int N = in_sizes[0] / IMG;       // 2048 samples
    dim3 grid(PNUM, N / 16);               // 128 patches x 128 sample tiles
    dim3 block(32);                        // one wave32 per block
    lca_wmma_kernel<<<grid, block, 0, stream>>>(x, We, be, Wd, bd, out);
}